// VoxelArtGrid_33749853012386
// MI455X (gfx1250) — compile-verified
//
#include <hip/hip_runtime.h>

// ---------------------------------------------------------------------------
// VoxelArtGrid forward: trilinear sample of |densities| (2ch) + features (6ch)
// on a 128^3 grid, hard-argmax both (straight-through softmax forward == hard
// one-hot), density = 1000*[argmax_d==1], rgb = palette[argmax_f].
//
// Memory-bound gather kernel. Roofline: ~176 MB mandatory HBM traffic
// (~7.6 us @ 23.3 TB/s); the 64 MB of volume data lives in the 192 MB L2, so
// the ~1 GB of corner gathers are L2-served. No matrix math exists in the
// forward pass (rgb is a 6x3 table lookup), so WMMA does not apply; we use
// the CDNA5-specific async global->LDS path for the palette broadcast and NT
// cache hints to keep L2 reserved for the volume.
// ---------------------------------------------------------------------------

#define VG_GRID 128
#define VG_NVOX (VG_GRID * VG_GRID * VG_GRID)

typedef __attribute__((ext_vector_type(2))) float v2f;
typedef __attribute__((ext_vector_type(4))) float v4f;

#if defined(__gfx1250__) && \
    __has_builtin(__builtin_amdgcn_global_load_async_to_lds_b32) && \
    __has_builtin(__builtin_amdgcn_s_wait_asynccnt)
#define VG_ASYNC_PAL 1
#else
#define VG_ASYNC_PAL 0
#endif

// Pack |densities| (2ch) + features (6ch) into one interleaved 8ch volume so
// the main kernel does two aligned b128 gathers per corner instead of four
// b64s, with fabs hoisted out of the 8-corner loop.
__global__ void vg_pack_kernel(const float* __restrict__ dens,
                               const float* __restrict__ feat,
                               float* __restrict__ packed) {
  int v = blockIdx.x * blockDim.x + threadIdx.x;
  if (v >= VG_NVOX) return;
  v2f dd = *(const v2f*)(dens + (size_t)v * 2);
  const v2f* fq = (const v2f*)(feat + (size_t)v * 6);
  v2f a = fq[0], b = fq[1], c = fq[2];
  v4f o0 = { __builtin_fabsf(dd.x), __builtin_fabsf(dd.y), a.x, a.y };
  v4f o1 = { b.x, b.y, c.x, c.y };
  v4f* o = (v4f*)(packed + (size_t)v * 8);
  o[0] = o0;
  o[1] = o1;
}

template <bool PACKED>
__global__ __launch_bounds__(256)
void vg_main_kernel(const float* __restrict__ points,
                    const float* __restrict__ dens,
                    const float* __restrict__ feat,
                    const float* __restrict__ packed,
                    const float* __restrict__ palette,
                    float* __restrict__ out, int n) {
  __shared__ float spal[18];

  // Broadcast the 72-byte palette into LDS (CDNA5 async global->LDS when the
  // toolchain exposes the builtin; plain load otherwise).
  unsigned t = threadIdx.x;
  if (t < 18u) {
#if VG_ASYNC_PAL
    __builtin_amdgcn_global_load_async_to_lds_b32(
        (__attribute__((address_space(1))) int*)(palette + t),
        (__attribute__((address_space(3))) int*)&spal[t],
        /*offset=*/0, /*cpol=*/0);
#else
    spal[t] = palette[t];
#endif
  }
#if VG_ASYNC_PAL
  __builtin_amdgcn_s_wait_asynccnt(0);
#endif
  __syncthreads();

  int i = blockIdx.x * blockDim.x + threadIdx.x;
  if (i >= n) return;

  // Points are read-once (48 MB): NT loads keep L2 for the hot volume.
  const float* pp = points + (size_t)i * 3;
  float px = __builtin_nontemporal_load(pp + 0);
  float py = __builtin_nontemporal_load(pp + 1);
  float pz = __builtin_nontemporal_load(pp + 2);

  // Reference: pos = ((p/64 + 1) * 128 - 1) * 0.5  (== p + 63.5)
  px = ((px * (1.0f / 64.0f) + 1.0f) * 128.0f - 1.0f) * 0.5f;
  py = ((py * (1.0f / 64.0f) + 1.0f) * 128.0f - 1.0f) * 0.5f;
  pz = ((pz * (1.0f / 64.0f) + 1.0f) * 128.0f - 1.0f) * 0.5f;

  float bx = floorf(px), by = floorf(py), bz = floorf(pz);
  float fx = px - bx, fy = py - by, fz = pz - bz;
  int ix = (int)bx, iy = (int)by, iz = (int)bz;
  // Inputs are strictly interior (base in [3,123]); clamp is just a guard.
  ix = min(max(ix, 0), VG_GRID - 2);
  iy = min(max(iy, 0), VG_GRID - 2);
  iz = min(max(iz, 0), VG_GRID - 2);

  float wx[2] = { 1.0f - fx, fx };
  float wy[2] = { 1.0f - fy, fy };
  float wz[2] = { 1.0f - fz, fz };

  int base = (ix * VG_GRID + iy) * VG_GRID + iz;

  float a0 = 0.f, a1 = 0.f, a2 = 0.f, a3 = 0.f;
  float a4 = 0.f, a5 = 0.f, a6 = 0.f, a7 = 0.f;

#pragma unroll
  for (int cx = 0; cx < 2; ++cx) {
#pragma unroll
    for (int cy = 0; cy < 2; ++cy) {
      float wxy = wx[cx] * wy[cy];
#pragma unroll
      for (int cz = 0; cz < 2; ++cz) {
        float w = wxy * wz[cz];
        int off = base + cx * (VG_GRID * VG_GRID) + cy * VG_GRID + cz;
        if (PACKED) {
          const v4f* q = (const v4f*)(packed + (size_t)off * 8);
          v4f lo = q[0], hi = q[1];
          a0 += w * lo.x; a1 += w * lo.y; a2 += w * lo.z; a3 += w * lo.w;
          a4 += w * hi.x; a5 += w * hi.y; a6 += w * hi.z; a7 += w * hi.w;
        } else {
          v2f dd = *(const v2f*)(dens + (size_t)off * 2);
          a0 += w * __builtin_fabsf(dd.x);
          a1 += w * __builtin_fabsf(dd.y);
          const v2f* fq = (const v2f*)(feat + (size_t)off * 6);
          v2f fa = fq[0], fb = fq[1], fc = fq[2];
          a2 += w * fa.x; a3 += w * fa.y;
          a4 += w * fb.x; a5 += w * fb.y;
          a6 += w * fc.x; a7 += w * fc.y;
        }
      }
    }
  }

  // jnp.argmax picks the FIRST max -> strict '>' keeps earlier index.
  float density = (a1 > a0) ? 1000.0f : 0.0f;

  int best = 0;
  float bv = a2;
  if (a3 > bv) { bv = a3; best = 1; }
  if (a4 > bv) { bv = a4; best = 2; }
  if (a5 > bv) { bv = a5; best = 3; }
  if (a6 > bv) { bv = a6; best = 4; }
  if (a7 > bv) { bv = a7; best = 5; }

  float r = spal[best * 3 + 0];
  float g = spal[best * 3 + 1];
  float b = spal[best * 3 + 2];

  // Output is write-once streaming (64 MB): NT store, keep L2 for the volume.
  v4f res = { density, r, g, b };
  __builtin_nontemporal_store(res, (v4f*)(out + (size_t)i * 4));
}

extern "C" void kernel_launch(void* const* d_in, const int* in_sizes, int n_in,
                              void* d_out, int out_size, void* d_ws, size_t ws_size,
                              hipStream_t stream) {
  const float* points    = (const float*)d_in[0];
  const float* densities = (const float*)d_in[1];
  const float* features  = (const float*)d_in[2];
  const float* palette   = (const float*)d_in[3];
  float* out = (float*)d_out;

  int n = in_sizes[0] / 3;
  int blocks = (n + 255) / 256;

  const size_t packed_bytes = (size_t)VG_NVOX * 8 * sizeof(float);  // 64 MB
  if (ws_size >= packed_bytes) {
    float* packed = (float*)d_ws;
    vg_pack_kernel<<<(VG_NVOX + 255) / 256, 256, 0, stream>>>(densities, features, packed);
    vg_main_kernel<true><<<blocks, 256, 0, stream>>>(points, densities, features,
                                                     packed, palette, out, n);
  } else {
    vg_main_kernel<false><<<blocks, 256, 0, stream>>>(points, densities, features,
                                                      nullptr, palette, out, n);
  }
}